// QUAFM_62088047231684
// MI455X (gfx1250) — compile-verified
//
#include <hip/hip_runtime.h>
#include <math.h>

typedef __attribute__((ext_vector_type(2))) float v2f;
typedef __attribute__((ext_vector_type(8))) float v8f;

#define B_    128
#define C_    512
#define T_    1024
#define OUT_  256
#define NMAT  256          // 2*B_: matrices 0..127 = refined, 128..255 = uncertainty
#define CHUNK 32           // matrices processed per pass (3 * 32MB chunk buffers in ws)
#define KDEG  28           // Chebyshev degrees 0..27
#define NT32  (C_ / 32)                  // 16 super-tiles per dim
#define NTILE (NT32 * (NT32 + 1) / 2)    // 136 lower-triangular super-tiles
#define EPSN  1e-12f
#define EPSE  1e-6f
#define PI_F  3.14159265358979f

// map linear triangular tile id -> (rt, ct) with rt >= ct
__device__ __forceinline__ void qf_tri_map(int t, int& rt, int& ct) {
    int r = (int)((sqrtf(8.0f * (float)t + 1.0f) - 1.0f) * 0.5f);
    while ((r + 1) * (r + 2) / 2 <= t) ++r;
    while (r * (r + 1) / 2 > t) --r;
    rt = r;
    ct = t - r * (r + 1) / 2;
}

// ---------------------------------------------------------------------------
// zero accumulators (traces + Gershgorin bounds)
// ---------------------------------------------------------------------------
__global__ void qf_zero_kernel(float* __restrict__ traces, float* __restrict__ bound) {
    int i = blockIdx.x * 256 + threadIdx.x;
    if (i < NMAT * KDEG) traces[i] = 0.0f;
    if (i < NMAT)        bound[i]  = 0.0f;
}

// ---------------------------------------------------------------------------
// Batched Gram: G[m] = F[m] * F[m]^T (symmetric) — lower-triangular tiles only,
// f32 WMMA, one wave per 32x32 tile (2x2 accumulators), mirror-store transpose.
// ---------------------------------------------------------------------------
__global__ __launch_bounds__(32)
void qf_gram_kernel(const float* __restrict__ Fr, const float* __restrict__ Fu,
                    float* __restrict__ G, int chunkStart) {
    const int lane = threadIdx.x;
    int rt, ct;
    qf_tri_map(blockIdx.x, rt, ct);
    const int lm = blockIdx.y;
    const int m  = chunkStart + lm;
    const float* __restrict__ F = (m < B_) ? (Fr + (size_t)m * C_ * T_)
                                           : (Fu + (size_t)(m - B_) * C_ * T_);
    const int r0 = rt * 32, c0 = ct * 32;
    const int mrow  = lane & 15;           // M (or N) within 16
    const int khalf = (lane >> 4) << 1;    // lanes 16-31 hold K=2,3

    v8f acc[2][2] = {};
    for (int kb = 0; kb < T_; kb += 4) {
        v2f a[2], b[2];
#pragma unroll
        for (int i = 0; i < 2; ++i) {
            a[i] = *(const v2f*)(F + (size_t)(r0 + 16 * i + mrow) * T_ + kb + khalf);
            // B = F^T : B[k][n] = F[c0+n][kb+k]  (row-contiguous per lane)
            b[i] = *(const v2f*)(F + (size_t)(c0 + 16 * i + mrow) * T_ + kb + khalf);
        }
#pragma unroll
        for (int i = 0; i < 2; ++i)
#pragma unroll
            for (int j = 0; j < 2; ++j)
                acc[i][j] = __builtin_amdgcn_wmma_f32_16x16x4_f32(
                    false, a[i], false, b[j], (short)0, acc[i][j], false, false);
    }
    float* __restrict__ Gm = G + (size_t)lm * C_ * C_;
    const int nn  = lane & 15;
    const int mhi = (lane >> 4) * 8;
    const bool offdiag = (rt != ct);
#pragma unroll
    for (int i = 0; i < 2; ++i)
#pragma unroll
        for (int j = 0; j < 2; ++j)
#pragma unroll
            for (int v = 0; v < 8; ++v) {
                int row = r0 + 16 * i + v + mhi;
                int col = c0 + 16 * j + nn;
                float g = acc[i][j][v];
                Gm[(size_t)row * C_ + col] = g;
                if (offdiag) Gm[(size_t)col * C_ + row] = g;   // symmetric mirror
            }
}

// ---------------------------------------------------------------------------
// row norms from diag(G):  nrm = max(sqrt(G[c,c]), 1e-12)
// ---------------------------------------------------------------------------
__global__ void qf_norm_kernel(const float* __restrict__ G, float* __restrict__ nrm) {
    int i = blockIdx.x * 256 + threadIdx.x;           // 0 .. CHUNK*C_-1
    if (i >= CHUNK * C_) return;
    int lm = i / C_, c = i % C_;
    float g = G[(size_t)lm * C_ * C_ + (size_t)c * C_ + c];
    nrm[i] = fmaxf(sqrtf(fmaxf(g, 0.0f)), EPSN);
}

// ---------------------------------------------------------------------------
// rho = G / (n_c * n_d * T), in place; Gershgorin bound[m] = max_c sum_d |rho|
// ---------------------------------------------------------------------------
__global__ __launch_bounds__(256)
void qf_rho_kernel(float* __restrict__ G, const float* __restrict__ nrm,
                   float* __restrict__ bound, int chunkStart) {
    const int c  = blockIdx.x;
    const int lm = blockIdx.y;
    const int m  = chunkStart + lm;
    float* __restrict__ Gm = G + (size_t)lm * C_ * C_;
    const float* __restrict__ nm = nrm + lm * C_;
    const float inv_c = 1.0f / (nm[c] * (float)T_);
    float rs = 0.0f;
    for (int d = threadIdx.x; d < C_; d += 256) {
        float v = Gm[(size_t)c * C_ + d] * inv_c / nm[d];
        Gm[(size_t)c * C_ + d] = v;
        rs += fabsf(v);
    }
    __shared__ float sm[256];
    sm[threadIdx.x] = rs;
    __syncthreads();
    for (int off = 128; off; off >>= 1) {
        if (threadIdx.x < off) sm[threadIdx.x] += sm[threadIdx.x + off];
        __syncthreads();
    }
    if (threadIdx.x == 0)   // rho is PSD -> row sums >= 0, uint compare == float compare
        atomicMax((unsigned int*)&bound[m], __float_as_uint(sm[0]));
}

// ---------------------------------------------------------------------------
// Y = 2*rho/b - I (in place over rho); Tc = Y (=T1); Tp = I (=T0);
// traces[m][1] += diag(Y)
// ---------------------------------------------------------------------------
__global__ void qf_yinit_kernel(float* __restrict__ Y, float* __restrict__ Tc,
                                float* __restrict__ Tp, const float* __restrict__ bound,
                                float* __restrict__ traces, int chunkStart) {
    size_t idx = (size_t)blockIdx.x * 256 + threadIdx.x;
    int lm  = (int)(idx / ((size_t)C_ * C_));
    int rem = (int)(idx % ((size_t)C_ * C_));
    int c = rem / C_, d = rem % C_;
    int m = chunkStart + lm;
    float bb = fmaxf(bound[m], 1e-30f);
    float y  = 2.0f * Y[idx] / bb - (c == d ? 1.0f : 0.0f);
    Y[idx]  = y;
    Tc[idx] = y;
    Tp[idx] = (c == d) ? 1.0f : 0.0f;
    if (c == d) atomicAdd(&traces[m * KDEG + 1], y);
}

// ---------------------------------------------------------------------------
// Chebyshev step (batched f32 WMMA GEMM):  Tp <- 2*Y*Tc - Tp, lower-triangle
// tiles only (result is symmetric), mirror-stored; traces[m][k] += diag.
// B operand loaded row-contiguously via the symmetry of Tc.
// In-place safe: each element is read+written by exactly one lane; the mirror
// position is write-only (its owning tile is never launched).
// ---------------------------------------------------------------------------
__global__ __launch_bounds__(32)
void qf_cheb_kernel(const float* __restrict__ Y, const float* __restrict__ Tc,
                    float* __restrict__ Tp, float* __restrict__ traces,
                    int chunkStart, int k) {
    const int lane = threadIdx.x;
    int rt, ct;
    qf_tri_map(blockIdx.x, rt, ct);
    const int lm = blockIdx.y;
    const int m  = chunkStart + lm;
    const float* __restrict__ Ym  = Y  + (size_t)lm * C_ * C_;
    const float* __restrict__ Tcm = Tc + (size_t)lm * C_ * C_;
    float*       __restrict__ Tpm = Tp + (size_t)lm * C_ * C_;
    const int r0 = rt * 32, c0 = ct * 32;
    const int mrow  = lane & 15;
    const int khalf = (lane >> 4) << 1;

    v8f acc[2][2] = {};
    for (int kb = 0; kb < C_; kb += 4) {
        v2f a[2], b[2];
#pragma unroll
        for (int i = 0; i < 2; ++i) {
            a[i] = *(const v2f*)(Ym  + (size_t)(r0 + 16 * i + mrow) * C_ + kb + khalf);
            // B[k][n] = Tc[k][c0+n] = Tc[c0+n][k]  (Tc symmetric -> contiguous load)
            b[i] = *(const v2f*)(Tcm + (size_t)(c0 + 16 * i + mrow) * C_ + kb + khalf);
        }
#pragma unroll
        for (int i = 0; i < 2; ++i)
#pragma unroll
            for (int j = 0; j < 2; ++j)
                acc[i][j] = __builtin_amdgcn_wmma_f32_16x16x4_f32(
                    false, a[i], false, b[j], (short)0, acc[i][j], false, false);
    }
    const int nn  = lane & 15;
    const int mhi = (lane >> 4) * 8;
    const bool offdiag = (rt != ct);
#pragma unroll
    for (int i = 0; i < 2; ++i)
#pragma unroll
        for (int j = 0; j < 2; ++j)
#pragma unroll
            for (int v = 0; v < 8; ++v) {
                int row = r0 + 16 * i + v + mhi;
                int col = c0 + 16 * j + nn;
                size_t idx = (size_t)row * C_ + col;
                float d = 2.0f * acc[i][j][v] - Tpm[idx];
                Tpm[idx] = d;
                if (offdiag)            Tpm[(size_t)col * C_ + row] = d;  // mirror
                else if (row == col)    atomicAdd(&traces[m * KDEG + k], d);
            }
}

// ---------------------------------------------------------------------------
// mean over T for both tensors
// ---------------------------------------------------------------------------
__global__ __launch_bounds__(256)
void qf_mean_kernel(const float* __restrict__ Fr, const float* __restrict__ Fu,
                    float* __restrict__ meanR, float* __restrict__ meanU) {
    const float* __restrict__ F = blockIdx.y ? Fu : Fr;
    float* __restrict__ M = blockIdx.y ? meanU : meanR;
    const float* __restrict__ row = F + (size_t)blockIdx.x * T_;
    float s = 0.0f;
    for (int t = threadIdx.x; t < T_; t += 256) s += row[t];
    __shared__ float sm[256];
    sm[threadIdx.x] = s;
    __syncthreads();
    for (int off = 128; off; off >>= 1) {
        if (threadIdx.x < off) sm[threadIdx.x] += sm[threadIdx.x + off];
        __syncthreads();
    }
    if (threadIdx.x == 0) M[blockIdx.x] = sm[0] * (1.0f / (float)T_);
}

// ---------------------------------------------------------------------------
// proj = mean @ W^T  (tiny: 34 MFLOP)
// ---------------------------------------------------------------------------
__global__ __launch_bounds__(256)
void qf_proj_kernel(const float* __restrict__ meanR, const float* __restrict__ meanU,
                    const float* __restrict__ W,
                    float* __restrict__ projR, float* __restrict__ projU) {
    int idx = blockIdx.x * 256 + threadIdx.x;   // 0 .. B_*OUT_-1
    if (idx >= B_ * OUT_) return;
    int b = idx / OUT_, o = idx % OUT_;
    const float* __restrict__ M = blockIdx.y ? meanU : meanR;
    float* __restrict__ P = blockIdx.y ? projU : projR;
    float s = 0.0f;
    for (int c = 0; c < C_; ++c) s += M[b * C_ + c] * W[o * C_ + c];
    P[idx] = s;
}

// ---------------------------------------------------------------------------
// final: per-batch entropies from Chebyshev traces (device-side DCT for the
// per-matrix coefficients, since each matrix has its own spectral bound b),
// then entropy-weighted combine of the two projections.
// ---------------------------------------------------------------------------
__global__ __launch_bounds__(256)
void qf_final_kernel(const float* __restrict__ traces, const float* __restrict__ bound,
                     const float* __restrict__ projR, const float* __restrict__ projU,
                     float* __restrict__ out) {
    const int b   = blockIdx.x;
    const int tid = threadIdx.x;
    __shared__ float partial[2 * KDEG];
    __shared__ float ent[2];

    if (tid < 2 * KDEG) {
        int sel = tid / KDEG;
        int j   = tid % KDEG;
        int m   = sel * B_ + b;
        float bb = fmaxf(bound[m], 1e-30f);
        // c_j = (2/N) * sum_i f(x_i) cos(j*theta_i),  f(x) = clip(x)*ln(clip(x))
        float cj = 0.0f;
        for (int i = 0; i < KDEG; ++i) {
            float th = PI_F * ((float)i + 0.5f) / (float)KDEG;
            float x  = bb * 0.5f * (cosf(th) + 1.0f);
            float xc = fmaxf(x, EPSE);
            cj += xc * logf(xc) * cosf((float)j * th);
        }
        cj *= 2.0f / (float)KDEG;
        float tj = (j == 0) ? (float)C_ * 0.5f          // (c0/2) * t0, t0 = C
                            : traces[m * KDEG + j];
        partial[tid] = cj * tj;
    }
    __syncthreads();
    if (tid < 2) {
        float s = 0.0f;
        for (int j = 0; j < KDEG; ++j) s += partial[tid * KDEG + j];
        ent[tid] = -s;                                   // entropy = -tr f(rho)
    }
    __syncthreads();
    float er = ent[0], eu = ent[1];
    float total = er + eu + EPSE;
    float wr = 1.0f - er / total;
    float wu = 1.0f - eu / total;
    out[b * OUT_ + tid] = wr * projR[b * OUT_ + tid] + wu * projU[b * OUT_ + tid];
}

// ---------------------------------------------------------------------------
extern "C" void kernel_launch(void* const* d_in, const int* in_sizes, int n_in,
                              void* d_out, int out_size, void* d_ws, size_t ws_size,
                              hipStream_t stream) {
    const float* Fr = (const float*)d_in[0];   // [B, C, T]
    const float* Fu = (const float*)d_in[1];   // [B, C, T]
    const float* W  = (const float*)d_in[2];   // [OUT, C]
    float* out = (float*)d_out;                // [B, OUT]

    char* ws = (char*)d_ws;
    size_t off = 0;
    auto alloc = [&](size_t bytes) -> float* {
        float* p = (float*)(ws + off);
        off += (bytes + 255) & ~(size_t)255;
        return p;
    };
    const size_t MATB = (size_t)C_ * C_ * sizeof(float);   // 1 MB per matrix
    float* Ybuf   = alloc(CHUNK * MATB);                   // Gram -> rho -> Y
    float* TcBuf  = alloc(CHUNK * MATB);
    float* TpBuf  = alloc(CHUNK * MATB);
    float* nrm    = alloc(CHUNK * C_ * sizeof(float));
    float* traces = alloc(NMAT * KDEG * sizeof(float));
    float* bound  = alloc(NMAT * sizeof(float));
    float* meanR  = alloc((size_t)B_ * C_ * sizeof(float));
    float* meanU  = alloc((size_t)B_ * C_ * sizeof(float));
    float* projR  = alloc((size_t)B_ * OUT_ * sizeof(float));
    float* projU  = alloc((size_t)B_ * OUT_ * sizeof(float));
    (void)ws_size; (void)in_sizes; (void)n_in; (void)out_size;

    qf_zero_kernel<<<(NMAT * KDEG + 255) / 256, 256, 0, stream>>>(traces, bound);

    // mean + projection path (independent of entropy path)
    qf_mean_kernel<<<dim3(B_ * C_, 2), 256, 0, stream>>>(Fr, Fu, meanR, meanU);
    qf_proj_kernel<<<dim3((B_ * OUT_ + 255) / 256, 2), 256, 0, stream>>>(
        meanR, meanU, W, projR, projU);

    // entropy path: 8 chunks of 32 matrices
    for (int cs = 0; cs < NMAT; cs += CHUNK) {
        qf_gram_kernel<<<dim3(NTILE, CHUNK), 32, 0, stream>>>(Fr, Fu, Ybuf, cs);
        qf_norm_kernel<<<(CHUNK * C_ + 255) / 256, 256, 0, stream>>>(Ybuf, nrm);
        qf_rho_kernel<<<dim3(C_, CHUNK), 256, 0, stream>>>(Ybuf, nrm, bound, cs);
        qf_yinit_kernel<<<(int)((size_t)CHUNK * C_ * C_ / 256), 256, 0, stream>>>(
            Ybuf, TcBuf, TpBuf, bound, traces, cs);
        float* tc = TcBuf;
        float* tp = TpBuf;
        for (int k = 2; k < KDEG; ++k) {
            qf_cheb_kernel<<<dim3(NTILE, CHUNK), 32, 0, stream>>>(
                Ybuf, tc, tp, traces, cs, k);
            float* t = tc; tc = tp; tp = t;
        }
    }

    qf_final_kernel<<<B_, 256, 0, stream>>>(traces, bound, projR, projU, out);
}